// RSAGEConv_68092411510977
// MI455X (gfx1250) — compile-verified
//
#include <hip/hip_runtime.h>
#include <hip/hip_bf16.h>

// MI455X / gfx1250, wave32. fp32 WMMA path: V_WMMA_F32_16X16X4_F32.
typedef __attribute__((ext_vector_type(2))) float v2f;
typedef __attribute__((ext_vector_type(8))) float v8f;

// ---------------------------------------------------------------- utilities

__global__ __launch_bounds__(256) void rsage_zero_f(float* __restrict__ p, long n4) {
  long i = ((long)blockIdx.x * blockDim.x + threadIdx.x);
  if (i < n4) {
    float4 z; z.x = 0.f; z.y = 0.f; z.z = 0.f; z.w = 0.f;
    *(float4*)(p + i * 4) = z;
  }
}

// out[i] = sum_r in[r*S + i]
__global__ __launch_bounds__(256) void rsage_reduce_r(const float* __restrict__ in,
                                                      float* __restrict__ out, int S, int R) {
  int i = blockIdx.x * blockDim.x + threadIdx.x;
  if (i >= S) return;
  float s = 0.f;
  for (int r = 0; r < R; ++r) s += in[(long)r * S + i];
  out[i] = s;
}

// degree count: one thread per (relation, edge)
__global__ __launch_bounds__(256) void rsage_deg(const int* __restrict__ dst,
                                                 float* __restrict__ deg,
                                                 int E, int N, long total) {
  long t = (long)blockIdx.x * blockDim.x + threadIdx.x;
  if (t >= total) return;
  int r = (int)(t / E);
  int d = dst[t];
  atomicAdd(&deg[(long)r * N + d], 1.0f);
}

__global__ __launch_bounds__(256) void rsage_invdeg(float* __restrict__ p, long n) {
  long i = (long)blockIdx.x * blockDim.x + threadIdx.x;
  if (i < n) p[i] = 1.0f / fmaxf(p[i], 1.0f);
}

// edge-parallel scatter-add: thread = (edge, 4-float chunk); acc[dst] += X[src]
__global__ __launch_bounds__(256) void rsage_scatter(const float* __restrict__ X,
                                                     const int* __restrict__ src,
                                                     const int* __restrict__ dst,
                                                     float* __restrict__ acc,
                                                     int D, int lg_chunks, long total) {
  long t = (long)blockIdx.x * blockDim.x + threadIdx.x;
  if (t >= total) return;
  long e = t >> lg_chunks;
  int cc = (int)((t & ((1l << lg_chunks) - 1)) << 2);
  int s = src[e];
  int d = dst[e];
  const float4 v = *(const float4*)(X + (long)s * D + cc);
  float* p = acc + (long)d * D + cc;
  atomicAdd(p + 0, v.x);
  atomicAdd(p + 1, v.y);
  atomicAdd(p + 2, v.z);
  atomicAdd(p + 3, v.w);
}

// ---------------------------------------------------------------- WMMA GEMM
// C[N,M] (mode 0: = bias + scale.*A@B ; mode 1: += scale.*A@B), optional ReLU.
// One wave computes a 16x64 strip of C (4 WMMA tiles reusing one A fragment).
// ISA layouts (CDNA5 ISA):
//   A 16x4 f32 : lanes 0-15 row=l, V0=K0 V1=K1 ; lanes 16-31 row=l-16, V0=K2 V1=K3
//   B 4x16 f32 : V0: lanes0-15 K0, lanes16-31 K2 ; V1: K1 / K3, col = lane&15
//   C 16x16 f32: VGPR i: lanes0-15 row i, lanes16-31 row i+8, col = lane&15
__global__ __launch_bounds__(128)
void rsage_gemm_wmma(const float* __restrict__ A, const float* __restrict__ B,
                     float* __restrict__ C, const float* __restrict__ rowscale,
                     const float* __restrict__ bias,
                     int K, int M, int row_tiles, int col_groups,
                     int mode, int relu) {
  const int wave = threadIdx.x >> 5;
  const int lane = threadIdx.x & 31;
  const long wid = (long)blockIdx.x * 4 + wave;
  const int col_grp = (int)(wid % col_groups);
  const int row_tile = (int)(wid / col_groups);
  if (row_tile >= row_tiles) return;  // wave-uniform: EXEC stays all-1s

  const int lhalf = lane >> 4;
  const int l15 = lane & 15;
  const int row0 = row_tile * 16;
  const int col0 = col_grp * 64;

  v8f acc[4];
  if (mode == 0) {
#pragma unroll
    for (int j = 0; j < 4; ++j) {
      float bv = bias ? bias[col0 + j * 16 + l15] : 0.0f;
#pragma unroll
      for (int i = 0; i < 8; ++i) acc[j][i] = bv;
    }
  } else {
#pragma unroll
    for (int j = 0; j < 4; ++j)
#pragma unroll
      for (int i = 0; i < 8; ++i)
        acc[j][i] = C[(long)(row0 + i + lhalf * 8) * M + col0 + j * 16 + l15];
  }

  const float* arow = A + (long)(row0 + l15) * K + lhalf * 2;
  const float* bcol = B + col0 + l15;
  const float scale = rowscale ? rowscale[row0 + l15] : 1.0f;

#pragma unroll 4
  for (int k = 0; k < K; k += 4) {
    v2f a = *(const v2f*)(arow + k);  // 8B aligned: row base even, k+2*lhalf even
    a.x *= scale;
    a.y *= scale;
    const float* bk = bcol + (long)(k + lhalf * 2) * M;
#pragma unroll
    for (int j = 0; j < 4; ++j) {
      v2f b;
      b.x = bk[j * 16];
      b.y = bk[(long)M + j * 16];
      acc[j] = __builtin_amdgcn_wmma_f32_16x16x4_f32(
          /*neg_a=*/false, a, /*neg_b=*/false, b,
          /*c_mod=*/(short)0, acc[j], /*reuse_a=*/false, /*reuse_b=*/false);
    }
  }

#pragma unroll
  for (int j = 0; j < 4; ++j)
#pragma unroll
    for (int i = 0; i < 8; ++i) {
      float v = acc[j][i];
      if (relu) v = fmaxf(v, 0.0f);
      C[(long)(row0 + i + lhalf * 8) * M + col0 + j * 16 + l15] = v;
    }
}

// ---------------------------------------------------------------- host side

static inline void launch_gemm(const float* A, const float* B, float* C,
                               const float* rowscale, const float* bias,
                               int Nn, int K, int M, int mode, int relu,
                               hipStream_t stream) {
  int row_tiles = Nn / 16;      // 50000 = 3125 * 16, exact
  int col_groups = M / 64;      // 256 -> 4, 128 -> 2, exact
  long waves = (long)row_tiles * col_groups;
  int blocks = (int)((waves + 3) / 4);
  rsage_gemm_wmma<<<blocks, 128, 0, stream>>>(A, B, C, rowscale, bias,
                                              K, M, row_tiles, col_groups, mode, relu);
}

static inline void launch_zero(float* p, long n_floats, hipStream_t stream) {
  long n4 = n_floats / 4;  // all buffers are multiples of 4 floats
  rsage_zero_f<<<(int)((n4 + 255) / 256), 256, 0, stream>>>(p, n4);
}

extern "C" void kernel_launch(void* const* d_in, const int* in_sizes, int n_in,
                              void* d_out, int out_size, void* d_ws, size_t ws_size,
                              hipStream_t stream) {
  const float* x       = (const float*)d_in[0];  // [N,128]
  const int*   src     = (const int*)d_in[1];    // [R,E]
  const int*   dst     = (const int*)d_in[2];    // [R,E]
  const float* Wself1  = (const float*)d_in[3];  // [R,128,256]
  const float* Wneigh1 = (const float*)d_in[4];  // [R,128,256]
  const float* b1      = (const float*)d_in[5];  // [R,256]
  const float* Wself2  = (const float*)d_in[6];  // [R,256,128]
  const float* Wneigh2 = (const float*)d_in[7];  // [R,256,128]
  const float* b2      = (const float*)d_in[8];  // [R,128]
  float* out = (float*)d_out;                    // [N,128]
  float* ws = (float*)d_ws;

  const int N = 50000, R = 3, E = 600000, DIN = 128, DHID = 256, DOUT = 128;

  // Workspace layout (floats), ~99 MiB total.
  size_t off = 0;
  float* invdeg = ws + off;  off += (size_t)R * N;        // 150000 (multiple of 16)
  float* wsum1  = ws + off;  off += (size_t)DIN * DHID;   // 32768
  float* bsum1  = ws + off;  off += 256;
  float* wsum2  = ws + off;  off += (size_t)DHID * DOUT;  // 32768
  float* bsum2  = ws + off;  off += 128;
  float* h      = ws + off;  off += (size_t)N * DHID;     // 12.8M
  float* meanb  = ws + off;  off += (size_t)N * DHID;     // 12.8M

  // 1) Summed self-weights / biases (HeteroGraphConv aggregate='sum').
  rsage_reduce_r<<<(DIN * DHID + 255) / 256, 256, 0, stream>>>(Wself1, wsum1, DIN * DHID, R);
  rsage_reduce_r<<<1, 256, 0, stream>>>(b1, bsum1, DHID, R);
  rsage_reduce_r<<<(DHID * DOUT + 255) / 256, 256, 0, stream>>>(Wself2, wsum2, DHID * DOUT, R);
  rsage_reduce_r<<<1, 128, 0, stream>>>(b2, bsum2, DOUT, R);

  // 2) Per-relation inverse degrees (shared by both layers).
  launch_zero(invdeg, (long)R * N, stream);
  {
    long total = (long)R * E;
    rsage_deg<<<(int)((total + 255) / 256), 256, 0, stream>>>(dst, invdeg, E, N, total);
    rsage_invdeg<<<(int)(((long)R * N + 255) / 256), 256, 0, stream>>>(invdeg, (long)R * N);
  }

  // 3) Layer 1: h = relu(x@Wsum1 + sum_r mean_r@Wneigh1_r + bsum1)
  launch_gemm(x, wsum1, h, nullptr, bsum1, N, DIN, DHID, /*mode=*/0, /*relu=*/0, stream);
  for (int r = 0; r < R; ++r) {
    launch_zero(meanb, (long)N * DIN, stream);
    long total = (long)E * (DIN / 4);
    rsage_scatter<<<(int)((total + 255) / 256), 256, 0, stream>>>(
        x, src + (long)r * E, dst + (long)r * E, meanb, DIN, /*lg_chunks=*/5, total);
    launch_gemm(meanb, Wneigh1 + (size_t)r * DIN * DHID, h, invdeg + (size_t)r * N,
                nullptr, N, DIN, DHID, /*mode=*/1, /*relu=*/(r == R - 1) ? 1 : 0, stream);
  }

  // 4) Layer 2: out = h@Wsum2 + sum_r mean_r@Wneigh2_r + bsum2
  launch_gemm(h, wsum2, out, nullptr, bsum2, N, DHID, DOUT, /*mode=*/0, /*relu=*/0, stream);
  for (int r = 0; r < R; ++r) {
    launch_zero(meanb, (long)N * DHID, stream);
    long total = (long)E * (DHID / 4);
    rsage_scatter<<<(int)((total + 255) / 256), 256, 0, stream>>>(
        h, src + (long)r * E, dst + (long)r * E, meanb, DHID, /*lg_chunks=*/6, total);
    launch_gemm(meanb, Wneigh2 + (size_t)r * DHID * DOUT, out, invdeg + (size_t)r * N,
                nullptr, N, DHID, DOUT, /*mode=*/1, /*relu=*/0, stream);
  }
}